// GIN_44624710205914
// MI455X (gfx1250) — compile-verified
//
#include <hip/hip_runtime.h>
#include <math.h>

typedef __attribute__((ext_vector_type(2))) float v2f;
typedef __attribute__((ext_vector_type(8))) float v8f;

// =====================================================================
// WMMA GEMM (fp32, 16x16x4): out[M,N] = maybe_relu(A[M,K] @ W[K,N] + bias)
// One wave per 16x16 output tile, 8 waves/block. K % 4 == 0, M % 16 == 0,
// N % 16 == 0 for all call sites. Weights staged TRANSPOSED in LDS with
// even padding so each B-fragment fetch is a single aligned ds_load_b64
// at a constant 16B stride per k-step (folds to immediate offsets).
// =====================================================================
__global__ __launch_bounds__(256) void gemm_wmma_f32(
    const float* __restrict__ A, const float* __restrict__ W,
    const float* __restrict__ bias, float* __restrict__ out,
    int M, int K, int N, int do_relu)
{
  __shared__ float s_wt[32 * 202];        // N * (K+2) max = 32*202 (25.9 KB)
  const int Kp = K + 2;                   // even pad: keeps 8B alignment, skews banks
  int tid = threadIdx.x;
  for (int i = tid; i < K * N; i += 256) {
    int kk = i / N, col = i - kk * N;
    s_wt[col * Kp + kk] = W[i];
  }
  __syncthreads();

  int lane = tid & 31;
  int wid  = tid >> 5;                    // 0..7
  int tiles_m = M >> 4;
  int tm = blockIdx.x * 8 + wid;          // m-tile index
  int tn = blockIdx.y;                    // n-tile index
  if (tm >= tiles_m) return;              // uniform per wave; EXEC stays all-1s
  int rbase = tm << 4;
  int cbase = tn << 4;

  int half = lane >> 4;                   // 0 = lanes 0-15, 1 = lanes 16-31
  int l16  = lane & 15;
  int arow = rbase + l16;                 // A: M striped over 16 lanes
  int bcol = cbase + l16;                 // B/C/D: N striped over 16 lanes

  // A 16x4 fp32 layout: lanes 0-15 -> {K=k,k+1}, lanes 16-31 -> {K=k+2,k+3}
  // B 4x16 fp32 layout mirrors A across the two lane halves.
  const float* a_ptr = A + (size_t)arow * K + (half << 1);
  const float* b_ptr = s_wt + bcol * Kp + (half << 1);

  v8f c = {};
  for (int k = 0; k < K; k += 4) {
    v2f a = *(const v2f*)(a_ptr + k);     // global_load_b64, stride 16B
    v2f b = *(const v2f*)(b_ptr + k);     // ds_load_b64,    stride 16B
    c = __builtin_amdgcn_wmma_f32_16x16x4_f32(
            /*neg_a=*/false, a, /*neg_b=*/false, b,
            /*c_mod=*/(short)0, c, /*reuse_a=*/false, /*reuse_b=*/false);
  }

  float bv = bias ? bias[bcol] : 0.0f;
  #pragma unroll
  for (int r = 0; r < 8; ++r) {
    int row = rbase + r + half * 8;       // C/D: VGPR r holds M=r (lo half), M=r+8 (hi half)
    float v = c[r] + bv;
    if (do_relu) v = fmaxf(v, 0.0f);
    out[(size_t)row * N + bcol] = v;
  }
}

// =====================================================================
// Per-graph edge aggregation in LDS (one block per graph):
//   hA = relu( y + segment_sum(y[src]) + bias )
// Each lane loads a distinct edge (coalesced), the wave broadcasts each
// edge via __shfl and all 32 lanes do the feature-wide ds_add_f32.
// valid==nullptr means every edge is valid. Node features are DIM=32.
// epg must be a multiple of 256 (it is: 6400).
// =====================================================================
__global__ __launch_bounds__(256) void aggr_kernel(
    const float* __restrict__ y, const int* __restrict__ src,
    const int* __restrict__ dst, const int* __restrict__ valid,
    const float* __restrict__ bias, float* __restrict__ hA,
    int n, int epg)
{
  __shared__ float s_y[200 * 32];   // 25.6 KB
  __shared__ float s_a[200 * 32];   // 25.6 KB
  int g = blockIdx.x;
  int tid = threadIdx.x;

  const float* yg = y + (size_t)g * n * 32;
  for (int i = tid; i < n * 32; i += 256) { s_y[i] = yg[i]; s_a[i] = 0.0f; }
  __syncthreads();

  int lane = tid & 31, wid = tid >> 5;
  size_t ebase = (size_t)g * epg;
  for (int e0 = wid * 32; e0 < epg; e0 += 8 * 32) {
    size_t ei = ebase + e0 + lane;
    int sle = src[ei];                         // coalesced: 32 edges per wave
    int dle = dst[ei];
    int vle = valid ? valid[ei] : 1;
    #pragma unroll 4
    for (int j = 0; j < 32; ++j) {
      if (!__shfl(vle, j, 32)) continue;
      int sl = __shfl(sle, j, 32) - g * n;
      int dl = __shfl(dle, j, 32) - g * n;
      atomicAdd(&s_a[dl * 32 + lane], s_y[sl * 32 + lane]);
    }
  }
  __syncthreads();

  float* hg = hA + (size_t)g * n * 32;
  for (int i = tid; i < n * 32; i += 256) {
    float v = s_y[i] + s_a[i] + bias[i & 31];
    hg[i] = fmaxf(v, 0.0f);
  }
}

// =====================================================================
// TopK pooling per graph (one block per graph):
//  score = tanh((h.pw)/||pw||); keep top-k (descending, stable);
//  hp = h[perm]*score[perm]; attn = sigmoid(score[perm]);
//  xcat[g, :32] (+)= max over kept rows, xcat[g, 32:64] (+)= mean;
//  optionally remap + validity-mask edges.
// =====================================================================
__global__ __launch_bounds__(256) void topk_pool_kernel(
    const float* __restrict__ h, const float* __restrict__ pw,
    float* __restrict__ hp, float* __restrict__ attn,
    float* __restrict__ xcat, int accum,
    int n, int k,
    const int* __restrict__ src_in, const int* __restrict__ dst_in,
    const int* __restrict__ valid_in,
    int* __restrict__ src_out, int* __restrict__ dst_out,
    int* __restrict__ valid_out, int epg, int do_edges)
{
  __shared__ float s_score[200];
  __shared__ int   s_newid[200];
  __shared__ float s_hp[100 * 32];
  int g = blockIdx.x, tid = threadIdx.x;

  float nrm = 0.0f;
  for (int i = 0; i < 32; ++i) nrm += pw[i] * pw[i];
  nrm = sqrtf(nrm);

  for (int i = tid; i < n; i += 256) {
    const float* row = h + ((size_t)g * n + i) * 32;
    float d = 0.0f;
    for (int f = 0; f < 32; ++f) d += row[f] * pw[f];
    s_score[i] = tanhf(d / nrm);
  }
  __syncthreads();

  for (int i = tid; i < n; i += 256) {
    float si = s_score[i];
    int r = 0;
    for (int j = 0; j < n; ++j) {
      float sj = s_score[j];
      if (sj > si || (sj == si && j < i)) ++r;   // descending, stable
    }
    int nid = (r < k) ? r : -1;
    s_newid[i] = nid;
    if (nid >= 0) {
      const float* row = h + ((size_t)g * n + i) * 32;
      for (int f = 0; f < 32; ++f) s_hp[nid * 32 + f] = row[f] * si;
      attn[(size_t)g * k + nid] = 1.0f / (1.0f + expf(-si));
    }
  }
  __syncthreads();

  if (hp) {
    float* hpg = hp + (size_t)g * k * 32;
    for (int i = tid; i < k * 32; i += 256) hpg[i] = s_hp[i];
  }
  if (tid < 32) {
    float mx = -INFINITY, sm = 0.0f;
    for (int r = 0; r < k; ++r) {
      float v = s_hp[r * 32 + tid];
      mx = fmaxf(mx, v);
      sm += v;
    }
    float mean = sm / (float)k;
    if (accum) {
      xcat[(size_t)g * 64 + tid] += mx;
      xcat[(size_t)g * 64 + 32 + tid] += mean;
    } else {
      xcat[(size_t)g * 64 + tid] = mx;
      xcat[(size_t)g * 64 + 32 + tid] = mean;
    }
  }
  if (do_edges) {
    size_t ebase = (size_t)g * epg;
    for (int e = tid; e < epg; e += 256) {
      size_t ei = ebase + e;
      int sl = src_in[ei] - g * n;
      int dl = dst_in[ei] - g * n;
      int vi = valid_in ? valid_in[ei] : 1;
      int ns = s_newid[sl], nd = s_newid[dl];
      int v = (vi && ns >= 0 && nd >= 0) ? 1 : 0;
      src_out[ei] = g * k + (ns > 0 ? ns : 0);
      dst_out[ei] = g * k + (nd > 0 ? nd : 0);
      valid_out[ei] = v;
    }
  }
}

// =====================================================================
// MLP head per graph: z(64) -> fc1(128)+relu+bn1 -> fc2(256)+relu+bn2
//                          -> fc3(2) -> log_softmax
// =====================================================================
__global__ __launch_bounds__(256) void mlp_head_kernel(
    const float* __restrict__ xsum,
    const float* __restrict__ fc1w, const float* __restrict__ fc1b,
    const float* __restrict__ fc2w, const float* __restrict__ fc2b,
    const float* __restrict__ fc3w, const float* __restrict__ fc3b,
    const float* __restrict__ bn1g, const float* __restrict__ bn1b,
    const float* __restrict__ bn1m, const float* __restrict__ bn1v,
    const float* __restrict__ bn2g, const float* __restrict__ bn2b,
    const float* __restrict__ bn2m, const float* __restrict__ bn2v,
    float* __restrict__ out_xy)
{
  __shared__ float s_z[64], s_h1[128], s_h2[256];
  int g = blockIdx.x, tid = threadIdx.x;

  if (tid < 64) s_z[tid] = xsum[(size_t)g * 64 + tid];
  __syncthreads();

  if (tid < 128) {
    float a = fc1b[tid];
    for (int i = 0; i < 64; ++i) a += s_z[i] * fc1w[i * 128 + tid];
    a = fmaxf(a, 0.0f);
    a = (a - bn1m[tid]) * rsqrtf(bn1v[tid] + 1e-5f) * bn1g[tid] + bn1b[tid];
    s_h1[tid] = a;
  }
  __syncthreads();

  {
    float a = fc2b[tid];
    for (int i = 0; i < 128; ++i) a += s_h1[i] * fc2w[i * 256 + tid];
    a = fmaxf(a, 0.0f);
    a = (a - bn2m[tid]) * rsqrtf(bn2v[tid] + 1e-5f) * bn2g[tid] + bn2b[tid];
    s_h2[tid] = a;
  }
  __syncthreads();

  if (tid == 0) {
    float v0 = fc3b[0], v1 = fc3b[1];
    for (int i = 0; i < 256; ++i) {
      v0 += s_h2[i] * fc3w[i * 2 + 0];
      v1 += s_h2[i] * fc3w[i * 2 + 1];
    }
    float m = fmaxf(v0, v1);
    float ls = m + logf(expf(v0 - m) + expf(v1 - m));
    out_xy[(size_t)g * 2 + 0] = v0 - ls;
    out_xy[(size_t)g * 2 + 1] = v1 - ls;
  }
}

// =====================================================================
extern "C" void kernel_launch(void* const* d_in, const int* in_sizes, int n_in,
                              void* d_out, int out_size, void* d_ws, size_t ws_size,
                              hipStream_t stream) {
  (void)in_sizes; (void)n_in; (void)out_size; (void)ws_size;

  const int B = 128, N = 200, F_IN = 200, DIM = 32;
  const int E = B * 6400;            // 819200 edges
  const int K1 = 100, K2 = 50;
  const int NODES1 = B * N;          // 25600
  const int NODES2 = B * K1;         // 12800

  // ---- inputs (setup_inputs dict order) ----
  const float* x    = (const float*)d_in[0];
  const int*   eidx = (const int*)  d_in[1];     // [2, E]
  // d_in[2] = batch (unused; implied by layout)
  const float* w1a  = (const float*)d_in[3];
  const float* b1a  = (const float*)d_in[4];
  const float* w1b  = (const float*)d_in[5];
  const float* b1b  = (const float*)d_in[6];
  const float* w2a  = (const float*)d_in[7];
  const float* b2a  = (const float*)d_in[8];
  const float* w2b  = (const float*)d_in[9];
  const float* b2b  = (const float*)d_in[10];
  const float* pw1  = (const float*)d_in[11];
  const float* pw2  = (const float*)d_in[12];
  const float* fc1w = (const float*)d_in[13];
  const float* fc1b = (const float*)d_in[14];
  const float* fc2w = (const float*)d_in[15];
  const float* fc2b = (const float*)d_in[16];
  const float* fc3w = (const float*)d_in[17];
  const float* fc3b = (const float*)d_in[18];
  const float* bn1g = (const float*)d_in[19];
  const float* bn1b = (const float*)d_in[20];
  const float* bn1m = (const float*)d_in[21];
  const float* bn1v = (const float*)d_in[22];
  const float* bn2g = (const float*)d_in[23];
  const float* bn2b = (const float*)d_in[24];
  const float* bn2m = (const float*)d_in[25];
  const float* bn2v = (const float*)d_in[26];

  const int* src = eidx;
  const int* dst = eidx + E;

  // ---- outputs: xy[256] ++ sigmoid(attn1)[12800] ++ sigmoid(attn2)[6400] ----
  float* out_xy = (float*)d_out;
  float* out_a1 = out_xy + B * 2;
  float* out_a2 = out_a1 + B * K1;

  // ---- workspace layout ----
  float* ws   = (float*)d_ws;
  float* bufA = ws;                       // NODES1*DIM = 819200 floats
  float* bufB = bufA + NODES1 * DIM;      // 819200
  float* bufC = bufB + NODES1 * DIM;      // 819200
  float* hp1  = bufC + NODES1 * DIM;      // NODES2*DIM = 409600
  float* hp2  = hp1  + NODES2 * DIM;      // B*K2*DIM   = 204800
  float* xsum = hp2  + B * K2 * DIM;      // B*64       = 8192
  int*   es2  = (int*)(xsum + B * 64);    // E
  int*   ed2  = es2 + E;                  // E
  int*   ev2  = ed2 + E;                  // E

  dim3 blk(256);

  // 1) y1 = x @ w1a                        [25600,200]x[200,32]  (WMMA)
  gemm_wmma_f32<<<dim3((NODES1 / 16) / 8, DIM / 16), blk, 0, stream>>>(
      x, w1a, nullptr, bufA, NODES1, F_IN, DIM, 0);

  // 2) hA = relu(y1 + segsum(y1[src]) + b1a)    (LDS edge aggregation)
  aggr_kernel<<<B, blk, 0, stream>>>(bufA, src, dst, nullptr, b1a, bufB, N, 6400);

  // 3) h1 = relu(hA @ w1b + b1b)           [25600,32]x[32,32]    (WMMA)
  gemm_wmma_f32<<<dim3((NODES1 / 16) / 8, DIM / 16), blk, 0, stream>>>(
      bufB, w1b, b1b, bufC, NODES1, DIM, DIM, 1);

  // 4) topk pool 1: hp1, attn1, x1 -> xsum, remapped edges
  topk_pool_kernel<<<B, blk, 0, stream>>>(
      bufC, pw1, hp1, out_a1, xsum, /*accum=*/0, N, K1,
      src, dst, nullptr, es2, ed2, ev2, 6400, /*do_edges=*/1);

  // 5) y2 = hp1 @ w2a                      [12800,32]x[32,32]    (WMMA)
  gemm_wmma_f32<<<dim3((NODES2 / 16) / 8, DIM / 16), blk, 0, stream>>>(
      hp1, w2a, nullptr, bufA, NODES2, DIM, DIM, 0);

  // 6) hA2 = relu(y2 + masked segsum + b2a)
  aggr_kernel<<<B, blk, 0, stream>>>(bufA, es2, ed2, ev2, b2a, bufB, K1, 6400);

  // 7) h2 = relu(hA2 @ w2b + b2b)          [12800,32]x[32,32]    (WMMA)
  gemm_wmma_f32<<<dim3((NODES2 / 16) / 8, DIM / 16), blk, 0, stream>>>(
      bufB, w2b, b2b, bufC, NODES2, DIM, DIM, 1);

  // 8) topk pool 2: attn2, x2 accumulated into xsum (no edges needed after)
  topk_pool_kernel<<<B, blk, 0, stream>>>(
      bufC, pw2, hp2, out_a2, xsum, /*accum=*/1, K1, K2,
      nullptr, nullptr, nullptr, nullptr, nullptr, nullptr, 6400, /*do_edges=*/0);

  // 9) MLP head + log_softmax
  mlp_head_kernel<<<B, blk, 0, stream>>>(
      xsum, fc1w, fc1b, fc2w, fc2b, fc3w, fc3b,
      bn1g, bn1b, bn1m, bn1v, bn2g, bn2b, bn2m, bn2v, out_xy);
}